// GCN_89172111000348
// MI455X (gfx1250) — compile-verified
//
#include <hip/hip_runtime.h>

// ---------------------------------------------------------------------------
// GCNConv (sym-norm) + ReLU + LayerNorm + Linear + ReLU for MI455X (gfx1250)
//
// Sizes: N=100000 nodes, E=1600000 edges, IN=64, HID=64, OUT=128.
// fp32 end to end -> use V_WMMA_F32_16X16X4_F32 (wave32 matrix pipe, fp32
// precision identical to reference). K=64 => 16 wmma steps per 16x16 tile.
// Edge phase: h (25.6MB) and agg (25.6MB) both live in the 192MB L2, so
// gather + atomic scatter-add run at L2 rate; HBM floor is only ~130MB.
// ---------------------------------------------------------------------------

#define N_NODES 100000
#define N_EDGES 1600000
#define IN_F 64
#define HID 64
#define OUT_F 128
#define LN_EPS 1e-5f

typedef __attribute__((ext_vector_type(2))) float v2f;
typedef __attribute__((ext_vector_type(8))) float v8f;

// ---------------------------------------------------------------------------
// K0: deg = 1 (self loop contribution), agg = 0
// ---------------------------------------------------------------------------
__global__ void k_init(float* __restrict__ deg, float* __restrict__ agg) {
  long long i = (long long)blockIdx.x * blockDim.x + threadIdx.x;
  if (i < (long long)N_NODES) deg[i] = 1.0f;
  if (i < (long long)N_NODES * HID) agg[i] = 0.0f;
}

// ---------------------------------------------------------------------------
// K1: deg[dst] += 1 per edge
// ---------------------------------------------------------------------------
__global__ void k_degree(const long long* __restrict__ ei, float* __restrict__ deg) {
  int e = blockIdx.x * blockDim.x + threadIdx.x;
  if (e >= N_EDGES) return;
  long long d = ei[(long long)N_EDGES + e];
  atomicAdd(&deg[d], 1.0f);
}

// ---------------------------------------------------------------------------
// K2: dinv = rsqrt(deg)   (deg >= 1 always, matches reference where() path)
// ---------------------------------------------------------------------------
__global__ void k_dinv(const float* __restrict__ deg, float* __restrict__ dinv) {
  int i = blockIdx.x * blockDim.x + threadIdx.x;
  if (i < N_NODES) dinv[i] = rsqrtf(deg[i]);
}

// ---------------------------------------------------------------------------
// K3: h = x @ W_conv  via V_WMMA_F32_16X16X4_F32.
// One wave owns a 16-row tile and all 4 N-tiles (N=64).
// A 16x4 fp32 fragment layout: lane L holds row M=L%16, K-half = L/16:
//   a.x = A[M][k0 + 2*(L/16)], a.y = A[M][k0 + 2*(L/16) + 1]
// B 4x16 fragment mirrors with N=L%16. C/D: VGPR j of lane L holds
//   D[j + 8*(L/16)][L%16] of the 16x16 tile.
// ---------------------------------------------------------------------------
__global__ void __launch_bounds__(256) k_gemm1(const float* __restrict__ x,
                                               const float* __restrict__ W,
                                               float* __restrict__ h) {
  const int wave = threadIdx.x >> 5;
  const int lane = threadIdx.x & 31;
  const int l16  = lane & 15;
  const int kh   = lane >> 4;  // 0 or 1
  const int rowBase = (blockIdx.x * 8 + wave) * 16;
  if (rowBase >= N_NODES) return;  // wave-uniform: EXEC stays all-1s for WMMA

  const int row = rowBase + l16;
  const float* xr = x + (long long)row * IN_F;

  v8f acc0 = {}, acc1 = {}, acc2 = {}, acc3 = {};
#pragma unroll
  for (int k0 = 0; k0 < IN_F; k0 += 4) {
    const int kb = k0 + 2 * kh;
    v2f a; a.x = xr[kb]; a.y = xr[kb + 1];
    v2f b;
    b.x = W[kb * HID + (0 * 16 + l16)]; b.y = W[(kb + 1) * HID + (0 * 16 + l16)];
    acc0 = __builtin_amdgcn_wmma_f32_16x16x4_f32(false, a, false, b, (short)0, acc0, false, false);
    b.x = W[kb * HID + (1 * 16 + l16)]; b.y = W[(kb + 1) * HID + (1 * 16 + l16)];
    acc1 = __builtin_amdgcn_wmma_f32_16x16x4_f32(false, a, false, b, (short)0, acc1, false, false);
    b.x = W[kb * HID + (2 * 16 + l16)]; b.y = W[(kb + 1) * HID + (2 * 16 + l16)];
    acc2 = __builtin_amdgcn_wmma_f32_16x16x4_f32(false, a, false, b, (short)0, acc2, false, false);
    b.x = W[kb * HID + (3 * 16 + l16)]; b.y = W[(kb + 1) * HID + (3 * 16 + l16)];
    acc3 = __builtin_amdgcn_wmma_f32_16x16x4_f32(false, a, false, b, (short)0, acc3, false, false);
  }

  const int mBase = 8 * kh;
#pragma unroll
  for (int j = 0; j < 8; ++j) {
    float* hr = h + (long long)(rowBase + mBase + j) * HID;
    hr[0 * 16 + l16] = acc0[j];
    hr[1 * 16 + l16] = acc1[j];
    hr[2 * 16 + l16] = acc2[j];
    hr[3 * 16 + l16] = acc3[j];
  }
}

// ---------------------------------------------------------------------------
// K4: agg[dst] += h[src] * dinv[src]*dinv[dst]  (one thread per edge,
// float4 gathers, global_atomic_add_f32 scatter; all L2-resident)
// ---------------------------------------------------------------------------
__global__ void k_scatter(const long long* __restrict__ ei,
                          const float* __restrict__ h,
                          const float* __restrict__ dinv,
                          float* __restrict__ agg) {
  int e = blockIdx.x * blockDim.x + threadIdx.x;
  if (e >= N_EDGES) return;
  long long s = ei[e];
  long long d = ei[(long long)N_EDGES + e];
  float norm = dinv[s] * dinv[d];
  const float* hs = h + s * HID;
  float* ad = agg + d * HID;
#pragma unroll
  for (int f = 0; f < HID; f += 4) {
    float4 v = *(const float4*)(hs + f);
    atomicAdd(ad + f + 0, v.x * norm);
    atomicAdd(ad + f + 1, v.y * norm);
    atomicAdd(ad + f + 2, v.z * norm);
    atomicAdd(ad + f + 3, v.w * norm);
  }
}

// ---------------------------------------------------------------------------
// K5 (fused epilogue): t = agg + h*dinv^2 + b_conv; relu; layernorm -> LDS;
// then per-wave 16x128 WMMA GEMM vs W1 (K=64) + b1 + relu -> out.
// Block = 256 threads = 8 waves = 128 rows. LDS row stride padded to 68
// floats so wmma A-fragment reads (stride 4 banks per row) are conflict-free.
// ---------------------------------------------------------------------------
__global__ void __launch_bounds__(256) k_epilogue(
    const float* __restrict__ agg, const float* __restrict__ h,
    const float* __restrict__ dinv, const float* __restrict__ b_conv,
    const float* __restrict__ gamma, const float* __restrict__ beta,
    const float* __restrict__ W1, const float* __restrict__ b1,
    float* __restrict__ out) {
  __shared__ float s_a[128][HID + 4];

  const int blockRow = blockIdx.x * 128;
  const int tid = threadIdx.x;

  // ---- Stage 1: bias + relu + layernorm into LDS (128 active threads) ----
  if (tid < 128) {
    const int row = blockRow + tid;
    if (row < N_NODES) {
      const float di = dinv[row];
      const float selfn = di * di;
      const float* ar = agg + (long long)row * HID;
      const float* hr = h + (long long)row * HID;
      float sum = 0.f, sq = 0.f;
#pragma unroll 4
      for (int f = 0; f < HID; ++f) {
        float t = ar[f] + hr[f] * selfn + b_conv[f];
        t = fmaxf(t, 0.f);
        s_a[tid][f] = t;
        sum += t;
        sq += t * t;
      }
      const float mu = sum * (1.0f / HID);
      const float var = sq * (1.0f / HID) - mu * mu;
      const float rstd = rsqrtf(var + LN_EPS);
#pragma unroll 4
      for (int f = 0; f < HID; ++f)
        s_a[tid][f] = (s_a[tid][f] - mu) * rstd * gamma[f] + beta[f];
    } else {
#pragma unroll 4
      for (int f = 0; f < HID; ++f) s_a[tid][f] = 0.f;
    }
  }
  __syncthreads();

  // ---- Stage 2: WMMA GEMM vs W1 [64x128], one wave per 16-row tile ----
  const int wave = tid >> 5;
  const int lane = tid & 31;
  const int l16  = lane & 15;
  const int kh   = lane >> 4;
  const int tileRow = blockRow + wave * 16;
  if (tileRow >= N_NODES) return;  // wave-uniform (N_NODES % 16 == 0)

  const int lr = wave * 16 + l16;  // LDS row for this lane's A fragment
  v8f acc[8] = {};
#pragma unroll
  for (int k0 = 0; k0 < HID; k0 += 4) {
    const int kb = k0 + 2 * kh;
    v2f a; a.x = s_a[lr][kb]; a.y = s_a[lr][kb + 1];
#pragma unroll
    for (int nt = 0; nt < 8; ++nt) {
      const int n = nt * 16 + l16;
      v2f b; b.x = W1[kb * OUT_F + n]; b.y = W1[(kb + 1) * OUT_F + n];
      acc[nt] = __builtin_amdgcn_wmma_f32_16x16x4_f32(false, a, false, b, (short)0, acc[nt], false, false);
    }
  }

  const int mBase = 8 * kh;
#pragma unroll
  for (int nt = 0; nt < 8; ++nt) {
    const int col = nt * 16 + l16;
    const float bb = b1[col];
#pragma unroll
    for (int j = 0; j < 8; ++j) {
      const int grow = tileRow + mBase + j;
      out[(long long)grow * OUT_F + col] = fmaxf(acc[nt][j] + bb, 0.f);
    }
  }
}

// ---------------------------------------------------------------------------
// Launch
// ---------------------------------------------------------------------------
extern "C" void kernel_launch(void* const* d_in, const int* in_sizes, int n_in,
                              void* d_out, int out_size, void* d_ws, size_t ws_size,
                              hipStream_t stream) {
  const float* x        = (const float*)d_in[0];
  const long long* ei   = (const long long*)d_in[1];  // int64 per reference
  const float* W_conv   = (const float*)d_in[2];
  const float* b_conv   = (const float*)d_in[3];
  const float* ln_gamma = (const float*)d_in[4];
  const float* ln_beta  = (const float*)d_in[5];
  const float* W1       = (const float*)d_in[6];
  const float* b1       = (const float*)d_in[7];
  float* out = (float*)d_out;

  // Workspace: deg[N] | dinv[N] | h[N*64] | agg[N*64]  (~51.8 MB)
  float* deg  = (float*)d_ws;
  float* dinv = deg + N_NODES;
  float* h    = dinv + N_NODES;
  float* agg  = h + (size_t)N_NODES * HID;

  k_init<<<((long long)N_NODES * HID + 255) / 256, 256, 0, stream>>>(deg, agg);
  k_degree<<<(N_EDGES + 255) / 256, 256, 0, stream>>>(ei, deg);
  k_dinv<<<(N_NODES + 255) / 256, 256, 0, stream>>>(deg, dinv);
  // 8 waves/block, 16 rows/wave -> 128 rows per block
  k_gemm1<<<(N_NODES + 127) / 128, 256, 0, stream>>>(x, W_conv, h);
  k_scatter<<<(N_EDGES + 255) / 256, 256, 0, stream>>>(ei, h, dinv, agg);
  k_epilogue<<<(N_NODES + 127) / 128, 256, 0, stream>>>(
      agg, h, dinv, b_conv, ln_gamma, ln_beta, W1, b1, out);

  // Tuple output: append edge_index (2 x E int64) after h [N x 128] f32.
  hipMemcpyAsync(out + (size_t)N_NODES * OUT_F, ei,
                 (size_t)2 * N_EDGES * sizeof(long long),
                 hipMemcpyDeviceToDevice, stream);
}